// GatedTCNLayerFullyConditioned_37194416783576
// MI455X (gfx1250) — compile-verified
//
#include <hip/hip_runtime.h>
#include <hip/hip_bf16.h>
#include <math.h>

// ---------------------------------------------------------------------------
// GatedTCNLayerFullyConditioned for MI455X (gfx1250).
//   f32 WMMA (V_WMMA_F32_16X16X4_F32) for all three GEMMs, Tensor Data Mover
//   (TENSOR_LOAD_TO_LDS) for all LDS tile staging, fused conv+gate+residual.
// ---------------------------------------------------------------------------

typedef __attribute__((ext_vector_type(2))) float v2f;
typedef __attribute__((ext_vector_type(8))) float v8f;

// 8-arg VOP3P pattern: (neg_a, A, neg_b, B, c_mod, C, reuse_a, reuse_b)
#define WMMA_F32(A, B, C) \
  __builtin_amdgcn_wmma_f32_16x16x4_f32(false, (A), false, (B), (short)0, (C), false, false)

constexpr int BATCH  = 16;
constexpr int CH     = 64;          // in == out channels of z
constexpr int OC     = 128;         // 2*CH gated channels
constexpr int KW     = 3;           // kernel taps
constexpr int DILN   = 4;           // dilation
constexpr int TLEN   = 16384;
constexpr int COND   = 128;
constexpr int KCOLS  = CH * KW;     // 192 = GEMM K for the conv
constexpr int KERN_N = OC * KCOLS;  // 24576 per-batch dynamic-kernel elements
constexpr int TT     = 128;         // timesteps per workgroup
constexpr int PAD    = (KW - 1) * DILN;  // 8 causal left pad
constexpr int XCOLS  = TT + PAD;    // 136

// LDS row strides. kernT/resi_w strides are produced by TDM LDS-padding
// (power-of-2 DWORD intervals): 128+8 and 64+4. x tile is stored unpadded
// (rows of 136 floats are contiguous = stride 136).
constexpr int KS = 136;   // kernT tile [192 x 136] = 104,448 B (cols = channel o)
constexpr int XS = 136;   // x tile    [ 64 x 136] =  34,816 B
constexpr int ZS = 130;   // z tile    [ 64 x 130] =  33,280 B
constexpr int RS = 68;    // resi_w    [ 64 x  68] =  17,408 B

// x-tile column offset for GEMM row r = i*3 + k of the dilated-tap matrix
__device__ __forceinline__ int xs_off(int r) {
  return (r / KW) * XS + (r % KW) * DILN;
}

// ---------------------------------------------------------------------------
// Tensor Data Mover: 2D tile load, descriptor per CDNA5 ISA sec. 8.3-8.6.
// ---------------------------------------------------------------------------
#if defined(__gfx1250__) && __has_builtin(__builtin_amdgcn_tensor_load_to_lds)
#define HAS_TDM 1
#if __has_include(<hip/amd_detail/amd_gfx1250_TDM.h>)
#define TDM_6ARG 1   // therock-10.0 headers -> 6-arg builtin
#endif
#endif

#ifdef HAS_TDM
typedef __attribute__((ext_vector_type(4))) unsigned tdm_u4;
typedef __attribute__((ext_vector_type(8))) int      tdm_i8;
typedef __attribute__((ext_vector_type(4))) int      tdm_i4;

// Load a [tile_d1 rows x tile_d0 cols] tile of f32 (row stride `row_stride`
// elements in global) into LDS at lds_ptr. Optional LDS padding: pad_amount
// code (n -> n+1 DWORDs) inserted every 2^(pad_interval+1) DWORDs stored.
__device__ __forceinline__ void tdm_load_2d(void* lds_ptr, const void* gptr,
                                            unsigned tile_d0, unsigned tile_d1,
                                            unsigned tensor_d0, unsigned tensor_d1,
                                            unsigned row_stride, bool pad_en,
                                            unsigned pad_interval, unsigned pad_amount) {
  // Generic LDS pointer: low 32 bits are the workgroup-relative LDS byte addr.
  const unsigned lds_addr = (unsigned)(unsigned long long)lds_ptr;
  const unsigned long long ga = (unsigned long long)gptr;
  tdm_u4 g0;
  g0[0] = 1u;                                            // count=1, user desc
  g0[1] = lds_addr;                                      // lds_addr
  g0[2] = (unsigned)ga;                                  // global_addr[31:0]
  g0[3] = ((unsigned)(ga >> 32) & 0x01FFFFFFu)           // global_addr[56:32]
          | (2u << 30);                                  // type = 2 ("image")
  tdm_i8 g1;
  g1[0] = (int)((2u << 16)                               // data_size = 4B
                | (pad_en ? (1u << 20) : 0u)             // pad_enable
                | (pad_interval << 22) | (pad_amount << 25));
  g1[1] = (int)((tensor_d0 & 0xFFFFu) << 16);            // tensor_dim0[15:0]
  g1[2] = (int)((tensor_d0 >> 16) | ((tensor_d1 & 0xFFFFu) << 16));
  g1[3] = (int)((tensor_d1 >> 16) | (tile_d0 << 16));    // tile_dim0
  g1[4] = (int)tile_d1;                                  // tile_dim1 (dim2=0)
  g1[5] = (int)row_stride;                               // tensor_dim0_stride
  g1[6] = 0;
  g1[7] = 0;
  const tdm_i4 gz = {0, 0, 0, 0};
#ifdef TDM_6ARG
  const tdm_i8 gz8 = {0, 0, 0, 0, 0, 0, 0, 0};
  __builtin_amdgcn_tensor_load_to_lds(g0, g1, gz, gz, gz8, 0);
#else
  __builtin_amdgcn_tensor_load_to_lds(g0, g1, gz, gz, 0);
#endif
}
#endif  // HAS_TDM

// ---------------------------------------------------------------------------
// Stage 1: A = c [16 x 128] (M = batch), B = adapter_w [128 x 24576] or
// bias_w [128 x 128]. One wave per 16-column N tile; C preloaded with bias.
// The adapter result is stored TRANSPOSED per batch: kern_all[b][cc][o]
// (cc = i*3+k in 0..191, o in 0..127) so stage 2 can TDM it straight into
// a bank-conflict-free A-operand layout.
// ---------------------------------------------------------------------------
__global__ __launch_bounds__(128)
void adapter_gemm(const float* __restrict__ c,  const float* __restrict__ aw,
                  const float* __restrict__ ab, const float* __restrict__ bw,
                  const float* __restrict__ bbv,
                  float* __restrict__ kern_all, float* __restrict__ biasv) {
  const int lane = threadIdx.x & 31;
  const int wv   = threadIdx.x >> 5;
  int tile = blockIdx.x * 4 + wv;

  const int NT_AD = KERN_N / 16;  // 1536 adapter tiles, then 8 bias tiles
  const bool is_ad = (tile < NT_AD);
  const float* Bm; const float* bias; int ncols;
  if (is_ad) { Bm = aw; bias = ab;  ncols = KERN_N; }
  else { tile -= NT_AD; Bm = bw; bias = bbv; ncols = OC; }

  const int nrow = lane & 15;   // N within tile / A-fragment M row
  const int half = lane >> 4;   // K sub-phase (0: K+0/1, 1: K+2/3)
  const int col  = tile * 16 + nrow;

  v8f acc;
  const float binit = bias[col];            // C[m][n] = bias[n]  (all m)
#pragma unroll
  for (int r = 0; r < 8; ++r) acc[r] = binit;

#pragma unroll
  for (int kk = 0; kk < COND / 4; ++kk) {
    const int k0 = 4 * kk + 2 * half;
    v2f a, b;
    a.x = c[nrow * COND + k0];
    a.y = c[nrow * COND + k0 + 1];
    b.x = Bm[(size_t)k0 * ncols + col];
    b.y = Bm[(size_t)(k0 + 1) * ncols + col];
    acc = WMMA_F32(a, b, acc);
  }
  if (is_ad) {
    const int o = col / KCOLS, cc = col - o * KCOLS;   // col -> (o, i*3+k)
#pragma unroll
    for (int r = 0; r < 8; ++r)                        // M = r+8*half = batch
      kern_all[(size_t)(r + 8 * half) * KERN_N + cc * OC + o] = acc[r];
  } else {
#pragma unroll
    for (int r = 0; r < 8; ++r)
      biasv[(r + 8 * half) * OC + col] = acc[r];
  }
}

// ---------------------------------------------------------------------------
// Stage 2: fused conv + gate + residual. Grid: (TLEN/TT, BATCH), 256 threads.
// ---------------------------------------------------------------------------
__global__ __launch_bounds__(256)
void tcn_main(const float* __restrict__ x,     const float* __restrict__ kern_all,
              const float* __restrict__ biasv, const float* __restrict__ rw,
              const float* __restrict__ rb,    float* __restrict__ outp,
              float* __restrict__ zoutp) {
  __shared__ float kernTS[KCOLS * KS];  // A of conv GEMM, transposed: [cc][o]
  __shared__ float xS[CH * XS];         // x tile incl. causal pad
  __shared__ float zS[CH * ZS];         // gated activations for residual GEMM
  __shared__ float rwS[CH * RS];        // resi_w [o][c]

  const int tid  = threadIdx.x;
  const int lane = tid & 31;
  const int wv   = tid >> 5;
  const int b    = blockIdx.y;
  const int t0   = blockIdx.x * TT;

  const float* kg = kern_all + (size_t)b * KERN_N;   // kernT[b]: [192][128]
  const float* xb = x + (size_t)b * CH * TLEN;

#ifdef HAS_TDM
  // One wave issues all three DMA tile loads; TENSORcnt tracks completion,
  // the workgroup barrier publishes the LDS writes to the other waves.
  if (wv == 0) {
    // kernT: rows of 128 DWORDs, pad 8 DWORDs/row -> LDS stride 136
    tdm_load_2d(kernTS, kg, OC, KCOLS, OC, KCOLS, OC, true, 6, 7);
    if (t0 == 0) {
      // interior 128 cols; rows of 128 DWORDs padded by 8 -> stride 136;
      // the 8 left-pad cols are zero-filled by the whole block below.
      tdm_load_2d(xS + PAD, xb, TT, CH, TLEN, CH, TLEN, true, 6, 7);
    } else {
      // full 136-col tile, rows contiguous -> stride 136, no padding needed
      tdm_load_2d(xS, xb + (t0 - PAD), XCOLS, CH, TLEN, CH, TLEN, false, 0, 0);
    }
    // resi_w: rows of 64 DWORDs, pad 4 DWORDs/row -> LDS stride 68
    tdm_load_2d(rwS, rw, CH, CH, CH, CH, CH, true, 5, 3);
    __builtin_amdgcn_s_wait_tensorcnt(0);
  }
  if (t0 == 0) {   // causal zero pad, disjoint from the TDM-written bytes
    for (int idx = tid; idx < CH * PAD; idx += 256)
      xS[(idx >> 3) * XS + (idx & 7)] = 0.0f;
  }
#else
  // Fallback staging without TDM (identical LDS layouts).
  for (int idx = tid; idx < KERN_N; idx += 256) {
    const int cc = idx >> 7, o = idx & (OC - 1);
    kernTS[cc * KS + o] = kg[idx];
  }
  for (int idx = tid; idx < CH * CH; idx += 256)
    rwS[(idx >> 6) * RS + (idx & 63)] = rw[idx];
  for (int idx = tid; idx < CH * XCOLS; idx += 256) {
    const int i = idx / XCOLS, j = idx - i * XCOLS;
    const int gt = t0 - PAD + j;
    xS[i * XS + j] = (gt >= 0) ? xb[i * TLEN + gt] : 0.0f;
  }
#endif
  __syncthreads();

  const int nrow = lane & 15;
  const int half = lane >> 4;
  const float* bv = biasv + b * OC;

  // ---- phase B: dynamic dilated conv + conditioned bias + gate ----
  // 32 (m, nt) tile tasks; m indexes a PAIR of channel tiles (tanh o=16m..,
  // sigmoid o=64+16m..) sharing one B fragment.
#pragma unroll 1
  for (int task = wv; task < 32; task += 8) {
    const int m  = task & 3;
    const int nt = task >> 2;
    v8f acc_t, acc_s;
#pragma unroll
    for (int r = 0; r < 8; ++r) {                  // C[m][n] = biasv[channel]
      acc_t[r] = bv[m * 16 + r + 8 * half];
      acc_s[r] = bv[64 + m * 16 + r + 8 * half];
    }
    const float* ka0 = &kernTS[m * 16 + nrow];        // A col = channel o
    const float* ka1 = &kernTS[64 + m * 16 + nrow];
    const int tl = nt * 16 + nrow;
#pragma unroll
    for (int kk = 0; kk < KCOLS / 4; ++kk) {       // 48 k-steps, 96 WMMAs/task
      const int k0 = 4 * kk + 2 * half;
      v2f a0, a1, bb;
      a0.x = ka0[k0 * KS];       a0.y = ka0[(k0 + 1) * KS];
      a1.x = ka1[k0 * KS];       a1.y = ka1[(k0 + 1) * KS];
      bb.x = xS[xs_off(k0)     + tl];
      bb.y = xS[xs_off(k0 + 1) + tl];
      acc_t = WMMA_F32(a0, bb, acc_t);
      acc_s = WMMA_F32(a1, bb, acc_s);
    }
    const int tg = t0 + tl;
#pragma unroll
    for (int r = 0; r < 8; ++r) {
      const int ch = m * 16 + r + 8 * half;
      const float zv = tanhf(acc_t[r]) * (1.0f / (1.0f + __expf(-acc_s[r])));
      zS[ch * ZS + tl] = zv;                                      // for residual
      zoutp[((size_t)b * CH + ch) * TLEN + tg] = zv;              // output z
    }
  }
  __syncthreads();

  // ---- phase C: residual GEMM out = resi_w @ z + resi_b + x ----
#pragma unroll 1
  for (int task = wv; task < 32; task += 8) {
    const int m  = task & 3;
    const int nt = task >> 2;
    v8f acc;
#pragma unroll
    for (int r = 0; r < 8; ++r) acc[r] = rb[m * 16 + r + 8 * half];
    const float* ra = &rwS[(m * 16 + nrow) * RS];
    const int tl = nt * 16 + nrow;
#pragma unroll
    for (int kk = 0; kk < CH / 4; ++kk) {          // 16 k-steps
      const int k0 = 4 * kk + 2 * half;
      v2f a, bb;
      a.x = ra[k0];  a.y = ra[k0 + 1];
      bb.x = zS[k0 * ZS + tl];
      bb.y = zS[(k0 + 1) * ZS + tl];
      acc = WMMA_F32(a, bb, acc);
    }
    const int tg = t0 + tl;
#pragma unroll
    for (int r = 0; r < 8; ++r) {
      const int o = m * 16 + r + 8 * half;
      outp[((size_t)b * CH + o) * TLEN + tg] = acc[r] + xS[o * XS + PAD + tl];
    }
  }
}

// ---------------------------------------------------------------------------
extern "C" void kernel_launch(void* const* d_in, const int* in_sizes, int n_in,
                              void* d_out, int out_size, void* d_ws, size_t ws_size,
                              hipStream_t stream) {
  const float* x  = (const float*)d_in[0];
  const float* c  = (const float*)d_in[1];
  const float* aw = (const float*)d_in[2];
  const float* ab = (const float*)d_in[3];
  const float* bw = (const float*)d_in[4];
  const float* bb = (const float*)d_in[5];
  const float* rw = (const float*)d_in[6];
  const float* rb = (const float*)d_in[7];

  float* outp  = (float*)d_out;                         // out: [16,64,16384]
  float* zoutp = outp + (size_t)BATCH * CH * TLEN;      // z:   [16,64,16384]

  float* kern_all = (float*)d_ws;                       // [16][192][128] (transposed)
  float* biasv    = kern_all + (size_t)BATCH * KERN_N;  // [16, 128]

  adapter_gemm<<<(KERN_N / 16 + OC / 16) / 4, 128, 0, stream>>>(
      c, aw, ab, bw, bb, kern_all, biasv);
  tcn_main<<<dim3(TLEN / TT, BATCH), 256, 0, stream>>>(
      x, kern_all, biasv, rw, rb, outp, zoutp);
}